// AEModel_61263413510327
// MI455X (gfx1250) — compile-verified
//
#include <hip/hip_runtime.h>
#include <math.h>

typedef __attribute__((ext_vector_type(2))) float v2f;
typedef __attribute__((ext_vector_type(8))) float v8f;

#define EPI_BIAS      0
#define EPI_BIAS_RELU 1
#define EPI_DIST      2

// Wait for all outstanding async global<->LDS copies (ASYNCcnt == 0).
__device__ __forceinline__ void wait_async0() {
#if __has_builtin(__builtin_amdgcn_s_wait_asynccnt)
    __builtin_amdgcn_s_wait_asynccnt(0);
#else
    asm volatile("s_wait_asynccnt 0x0" ::: "memory");
#endif
}

// One 16-byte-per-lane async copy from the global address base + goff to the LDS address in lds.
// GLOBAL_LOAD_ASYNC_TO_LDS_B128, GVS mode (saddr base + 32-bit vaddr offset).
__device__ __forceinline__ void async_cp_b128(uint32_t lds, uint32_t goff,
                                              const void* base) {
    asm volatile("global_load_async_to_lds_b128 %0, %1, %2"
                 :
                 : "v"(lds), "v"(goff), "s"(base)
                 : "memory");
}

// ---------------------------------------------------------------------------
// Generic 128x128-tile fp32 WMMA GEMM, double-buffered LDS, async-DMA pipeline.
//   C[M,N] = A[M,K] @ B  (+ epilogue)
//   BT=false: B is [K,N] row-major.  BT=true: B is [N,K] row-major (A @ B^T).
// 256 threads = 8 waves (2 M-waves x 4 N-waves), wave tile 64x32,
// each wave holds 4x2 accumulators of v_wmma_f32_16x16x4_f32.
// Stage s+1 is copied global->LDS by the async engine (ASYNCcnt) while the
// wave runs the 32-WMMA burst of stage s; s_wait_asynccnt 0 + one barrier per
// stage. No VGPR staging -> no scratch spills.
// ---------------------------------------------------------------------------
template <int EPI, bool BT>
__global__ __launch_bounds__(256) void gemm_wmma(
    const float* __restrict__ A, const float* __restrict__ Bm,
    const float* __restrict__ bias,
    const float* __restrict__ a2, const float* __restrict__ c2,
    float* __restrict__ C, int M, int N, int K)
{
    constexpr int BMt = 128, BNt = 128, BK = 16;
    constexpr int LDA = BK + 4;                       // 20 floats, even -> b64-aligned frags
    constexpr int LDB = BT ? (BK + 4) : (BNt + 4);    // 20 or 132
    constexpr int SBN = BT ? (BMt * LDB) : (BK * LDB);

    __shared__ __align__(16) float sA[2][BMt * LDA];
    __shared__ __align__(16) float sB[2][SBN];

    const int tid  = threadIdx.x;
    const int lane = tid & 31;
    const int wave = tid >> 5;
    const int half = lane >> 4;   // 0: K=0/1 lanes, 1: K=2/3 lanes (16x4 f32 A layout)
    const int lm   = lane & 15;
    const int wM   = wave >> 2;   // 0..1
    const int wN   = wave & 3;    // 0..3

    const int rowBase = blockIdx.y * BMt;
    const int colBase = blockIdx.x * BNt;

    v8f acc[4][2] = {};

    // flat-id decompositions for the staging copies (2 x b128 per thread per matrix)
    const int arf0 = tid >> 2,          ac0 = (tid & 3) << 2;          // A / BT-B part 0
    const int arf1 = (tid + 256) >> 2,  ac1 = ((tid + 256) & 3) << 2;  // A / BT-B part 1
    const int brf0 = tid >> 5,          bc0 = (tid & 31) << 2;         // normal-B part 0
    const int brf1 = (tid + 256) >> 5,  bc1 = ((tid + 256) & 31) << 2; // normal-B part 1

    // LDS byte addresses (low 32 bits of the flat shared pointer = LDS offset)
    constexpr uint32_t ABYTES = (uint32_t)(BMt * LDA * 4);
    constexpr uint32_t BBYTES = (uint32_t)(SBN * 4);
    const uint32_t ldsA0 = (uint32_t)(uintptr_t)&sA[0][0] + (uint32_t)(arf0 * LDA + ac0) * 4u;
    const uint32_t ldsA1 = (uint32_t)(uintptr_t)&sA[0][0] + (uint32_t)(arf1 * LDA + ac1) * 4u;
    const uint32_t ldsB0 = (uint32_t)(uintptr_t)&sB[0][0] +
        (BT ? (uint32_t)(arf0 * LDB + ac0) : (uint32_t)(brf0 * LDB + bc0)) * 4u;
    const uint32_t ldsB1 = (uint32_t)(uintptr_t)&sB[0][0] +
        (BT ? (uint32_t)(arf1 * LDB + ac1) : (uint32_t)(brf1 * LDB + bc1)) * 4u;

    // global byte offsets at k = 0 (all operands are < 2^31 bytes)
    const uint32_t gA0 = (uint32_t)((rowBase + arf0) * K + ac0) * 4u;
    const uint32_t gA1 = (uint32_t)((rowBase + arf1) * K + ac1) * 4u;
    const uint32_t gB0 = BT ? (uint32_t)((colBase + arf0) * K + ac0) * 4u
                            : (uint32_t)(brf0 * N + colBase + bc0) * 4u;
    const uint32_t gB1 = BT ? (uint32_t)((colBase + arf1) * K + ac1) * 4u
                            : (uint32_t)(brf1 * N + colBase + bc1) * 4u;

    auto issue_stage = [&](int buf, int kn) {
        const uint32_t ab = (uint32_t)buf * ABYTES;
        const uint32_t bb = (uint32_t)buf * BBYTES;
        const uint32_t ka = (uint32_t)kn * 4u;          // k advance for A / BT-B
        async_cp_b128(ldsA0 + ab, gA0 + ka, A);
        async_cp_b128(ldsA1 + ab, gA1 + ka, A);
        if constexpr (BT) {
            async_cp_b128(ldsB0 + bb, gB0 + ka, Bm);
            async_cp_b128(ldsB1 + bb, gB1 + ka, Bm);
        } else {
            const uint32_t kb = (uint32_t)kn * (uint32_t)N * 4u;
            async_cp_b128(ldsB0 + bb, gB0 + kb, Bm);
            async_cp_b128(ldsB1 + bb, gB1 + kb, Bm);
        }
    };

    // --- pipeline prologue: DMA stage 0 into buffer 0 ---
    issue_stage(0, 0);
    wait_async0();
    __syncthreads();

    const int nstage = K / BK;
    for (int s = 0; s < nstage; ++s) {
        const int cur = s & 1;
        const bool hasNext = (s + 1) < nstage;

        // kick off the async DMA for stage s+1 into the other buffer
        if (hasNext) issue_stage(cur ^ 1, (s + 1) * BK);

        // --- 4 wmma k-steps of depth 4 on the current buffer ---
        const float* cA = sA[cur];
        const float* cB = sB[cur];
        #pragma unroll
        for (int kk = 0; kk < BK; kk += 4) {
            v2f af[4], bf[2];
            #pragma unroll
            for (int mi = 0; mi < 4; ++mi) {
                const int r = wM * 64 + mi * 16 + lm;
                af[mi] = *reinterpret_cast<const v2f*>(&cA[r * LDA + kk + half * 2]);
            }
            #pragma unroll
            for (int ni = 0; ni < 2; ++ni) {
                const int c = wN * 32 + ni * 16 + lm;
                if constexpr (BT) {
                    bf[ni] = *reinterpret_cast<const v2f*>(&cB[c * LDB + kk + half * 2]);
                } else {
                    bf[ni].x = cB[(kk + half * 2 + 0) * LDB + c];
                    bf[ni].y = cB[(kk + half * 2 + 1) * LDB + c];
                }
            }
            #pragma unroll
            for (int mi = 0; mi < 4; ++mi)
                #pragma unroll
                for (int ni = 0; ni < 2; ++ni)
                    acc[mi][ni] = __builtin_amdgcn_wmma_f32_16x16x4_f32(
                        false, af[mi], false, bf[ni], (short)0, acc[mi][ni],
                        false, false);
        }

        // drain this wave's DMA, then one barrier per stage
        if (hasNext) wait_async0();
        __syncthreads();
    }

    // --- epilogue ---
    #pragma unroll
    for (int mi = 0; mi < 4; ++mi) {
        #pragma unroll
        for (int ni = 0; ni < 2; ++ni) {
            const int n = colBase + wN * 32 + ni * 16 + lm;
            float bv = 0.0f, cn = 0.0f;
            if constexpr (EPI == EPI_DIST) cn = c2[n]; else bv = bias[n];
            #pragma unroll
            for (int v = 0; v < 8; ++v) {
                const int m = rowBase + wM * 64 + mi * 16 + half * 8 + v;
                float x = acc[mi][ni][v];
                if constexpr (EPI == EPI_DIST) {
                    x = 2.0f * x - a2[m] - cn;      // = -(||a||^2 - 2ab + ||c||^2)
                } else {
                    x += bv;
                    if constexpr (EPI == EPI_BIAS_RELU) x = fmaxf(x, 0.0f);
                }
                C[(size_t)m * N + n] = x;
            }
        }
    }
}

// ---------------------------------------------------------------------------
// BatchNorm training stats: per-column mean / rstd over M rows (biased var).
// ---------------------------------------------------------------------------
__global__ __launch_bounds__(256) void colstats(
    const float* __restrict__ Z, float* __restrict__ mean,
    float* __restrict__ rstd, int M, int Ncols)
{
    const int c = blockIdx.x * blockDim.x + threadIdx.x;
    if (c >= Ncols) return;
    float s = 0.0f, ss = 0.0f;
    for (int r = 0; r < M; ++r) {
        const float x = Z[(size_t)r * Ncols + c];
        s += x; ss += x * x;
    }
    const float m = s / (float)M;
    const float v = ss / (float)M - m * m;
    mean[c] = m;
    rstd[c] = rsqrtf(v + 1e-3f);
}

// ---------------------------------------------------------------------------
// In-place BN apply + ReLU (float4 vectorized; Ncols % 4 == 0).
// ---------------------------------------------------------------------------
__global__ __launch_bounds__(256) void bn_relu(
    float* __restrict__ Z, const float* __restrict__ mean,
    const float* __restrict__ rstd, const float* __restrict__ gamma,
    const float* __restrict__ beta, int total, int Ncols)
{
    const int i4 = blockIdx.x * blockDim.x + threadIdx.x;
    const int base = i4 * 4;
    if (base >= total) return;
    const int c = base % Ncols;
    float4 x = *reinterpret_cast<float4*>(&Z[base]);
    const float4 mu = *reinterpret_cast<const float4*>(&mean[c]);
    const float4 rs = *reinterpret_cast<const float4*>(&rstd[c]);
    const float4 g  = *reinterpret_cast<const float4*>(&gamma[c]);
    const float4 b  = *reinterpret_cast<const float4*>(&beta[c]);
    x.x = fmaxf(g.x * (x.x - mu.x) * rs.x + b.x, 0.0f);
    x.y = fmaxf(g.y * (x.y - mu.y) * rs.y + b.y, 0.0f);
    x.z = fmaxf(g.z * (x.z - mu.z) * rs.z + b.z, 0.0f);
    x.w = fmaxf(g.w * (x.w - mu.w) * rs.w + b.w, 0.0f);
    *reinterpret_cast<float4*>(&Z[base]) = x;
}

// ---------------------------------------------------------------------------
// Row squared norms for [rows, cols] (cols small, e.g. 128). One wave per row.
// ---------------------------------------------------------------------------
__global__ __launch_bounds__(256) void rownorm2(
    const float* __restrict__ X, float* __restrict__ out, int rows, int cols)
{
    const int warp = (int)((blockIdx.x * blockDim.x + threadIdx.x) >> 5);
    const int lane = threadIdx.x & 31;
    if (warp >= rows) return;
    const float* row = X + (size_t)warp * cols;
    float s = 0.0f;
    for (int j = lane; j < cols; j += 32) { const float v = row[j]; s += v * v; }
    #pragma unroll
    for (int off = 16; off > 0; off >>= 1) s += __shfl_down(s, off, 32);
    if (lane == 0) out[warp] = s;
}

// ---------------------------------------------------------------------------
// Per-row argmax of assignment (= argmin of dist, smallest index on ties),
// then gather the winning codebook row into vq_feat.
// ---------------------------------------------------------------------------
__global__ __launch_bounds__(256) void argmax_gather(
    const float* __restrict__ assign, const float* __restrict__ ctx,
    float* __restrict__ vq, int Kc, int dm)
{
    const int row = blockIdx.x;
    const float* a = assign + (size_t)row * Kc;
    float best = -__builtin_inff();
    int bidx = 0x7fffffff;
    for (int j = threadIdx.x; j < Kc; j += 256) {
        const float v = a[j];
        if (v > best || (v == best && j < bidx)) { best = v; bidx = j; }
    }
    __shared__ float sv[256];
    __shared__ int   si[256];
    sv[threadIdx.x] = best; si[threadIdx.x] = bidx;
    __syncthreads();
    for (int s = 128; s > 0; s >>= 1) {
        if ((int)threadIdx.x < s) {
            const float v2 = sv[threadIdx.x + s];
            const int   i2 = si[threadIdx.x + s];
            if (v2 > sv[threadIdx.x] ||
                (v2 == sv[threadIdx.x] && i2 < si[threadIdx.x])) {
                sv[threadIdx.x] = v2; si[threadIdx.x] = i2;
            }
        }
        __syncthreads();
    }
    const int bi = si[0];
    for (int j = threadIdx.x; j < dm; j += 256)
        vq[(size_t)row * dm + j] = ctx[(size_t)bi * dm + j];
}

// ---------------------------------------------------------------------------
extern "C" void kernel_launch(void* const* d_in, const int* in_sizes, int n_in,
                              void* d_out, int out_size, void* d_ws, size_t ws_size,
                              hipStream_t stream)
{
    (void)in_sizes; (void)n_in; (void)out_size; (void)ws_size;

    const float* x   = (const float*)d_in[0];
    const float* We1 = (const float*)d_in[1];
    const float* be1 = (const float*)d_in[2];
    const float* ge  = (const float*)d_in[3];
    const float* bbe = (const float*)d_in[4];
    const float* We2 = (const float*)d_in[5];
    const float* be2 = (const float*)d_in[6];
    const float* ctx = (const float*)d_in[7];
    const float* Wd1 = (const float*)d_in[8];
    const float* bd1 = (const float*)d_in[9];
    const float* gd  = (const float*)d_in[10];
    const float* bbd = (const float*)d_in[11];
    const float* Wd2 = (const float*)d_in[12];
    const float* bd2 = (const float*)d_in[13];

    constexpr int Bn = 4096, Dn = 4096, Hn = 2048, DM = 128, Kc = 4096;

    float* out    = (float*)d_out;
    float* pred   = out;                                  // [B, D]
    float* assign = out + (size_t)Bn * Dn;                // [B, K]
    float* feat   = assign + (size_t)Bn * Kc;             // [B, dm]

    float* ws   = (float*)d_ws;
    float* Z    = ws;                                     // [B, H] (reused enc/dec)
    float* vq   = Z + (size_t)Bn * Hn;                    // [B, dm]
    float* a2   = vq + (size_t)Bn * DM;                   // [B]
    float* c2v  = a2 + Bn;                                // [K]
    float* mean = c2v + Kc;                               // [H]
    float* rstd = mean + Hn;                              // [H]

    const dim3 blk(256);

    // --- encoder ---
    gemm_wmma<EPI_BIAS, false><<<dim3(Hn / 128, Bn / 128), blk, 0, stream>>>(
        x, We1, be1, nullptr, nullptr, Z, Bn, Hn, Dn);
    colstats<<<dim3(Hn / 256), blk, 0, stream>>>(Z, mean, rstd, Bn, Hn);
    bn_relu<<<dim3((Bn * Hn / 4 + 255) / 256), blk, 0, stream>>>(
        Z, mean, rstd, ge, bbe, Bn * Hn, Hn);
    gemm_wmma<EPI_BIAS, false><<<dim3(DM / 128, Bn / 128), blk, 0, stream>>>(
        Z, We2, be2, nullptr, nullptr, feat, Bn, DM, Hn);

    // --- VQ: assignment = -(||f||^2 - 2 f.c + ||c||^2), argmin -> gather ---
    rownorm2<<<dim3(Bn / 8), blk, 0, stream>>>(feat, a2, Bn, DM);
    rownorm2<<<dim3(Kc / 8), blk, 0, stream>>>(ctx, c2v, Kc, DM);
    gemm_wmma<EPI_DIST, true><<<dim3(Kc / 128, Bn / 128), blk, 0, stream>>>(
        feat, ctx, nullptr, a2, c2v, assign, Bn, Kc, DM);
    argmax_gather<<<dim3(Bn), blk, 0, stream>>>(assign, ctx, vq, Kc, DM);

    // --- decoder ---
    gemm_wmma<EPI_BIAS, false><<<dim3(Hn / 128, Bn / 128), blk, 0, stream>>>(
        vq, Wd1, bd1, nullptr, nullptr, Z, Bn, Hn, DM);
    colstats<<<dim3(Hn / 256), blk, 0, stream>>>(Z, mean, rstd, Bn, Hn);
    bn_relu<<<dim3((Bn * Hn / 4 + 255) / 256), blk, 0, stream>>>(
        Z, mean, rstd, gd, bbd, Bn * Hn, Hn);
    gemm_wmma<EPI_BIAS_RELU, false><<<dim3(Dn / 128, Bn / 128), blk, 0, stream>>>(
        Z, Wd2, bd2, nullptr, nullptr, pred, Bn, Dn, Hn);
}